// OFTTAPrototypeHead_67800353734667
// MI455X (gfx1250) — compile-verified
//
#include <hip/hip_runtime.h>
#include <math.h>

#define D_DIM 512
#define KMAX  10
#define EPSV  1e-12f

typedef float v2f __attribute__((ext_vector_type(2)));
typedef float v8f __attribute__((ext_vector_type(8)));

__device__ __forceinline__ v8f wmma4(v2f a, v2f b, v8f c) {
  // V_WMMA_F32_16X16X4_F32 : D = A(16x4) * B(4x16) + C(16x16)
  return __builtin_amdgcn_wmma_f32_16x16x4_f32(false, a, false, b, (short)0, c, false, false);
}

// Shared WMMA inner-product for one 16(row)x16(col) tile over K=512.
// A rows from `arow` (feat), B rows from `brow`. Pad columns (c >= C) are
// handled by clamping `brow` to a valid row at the call site; the resulting
// duplicate accumulator columns are never read, so no masking is needed and
// the loop body is pure loads + wmma (no VALU between load and consume).
__device__ __forceinline__ v8f tile_gemm(const float* __restrict__ arow,
                                         const float* __restrict__ brow) {
  v8f acc = {0.f,0.f,0.f,0.f,0.f,0.f,0.f,0.f};
#pragma unroll 8
  for (int k0 = 0; k0 < D_DIM; k0 += 4) {
    if ((k0 & 127) == 0) {
      // pull the next 1KB of the feat stream ahead of the consume loop
      __builtin_prefetch(arow + k0 + 256, 0, 3);   // -> global_prefetch_b8
    }
    v2f a, bf;
    a.x  = arow[k0]; a.y  = arow[k0 + 1];
    bf.x = brow[k0]; bf.y = brow[k0 + 1];
    acc = wmma4(a, bf, acc);
  }
  return acc;
}

// ---------------------------------------------------------------- stage A:
// wl = W@W^T + b (CxC); per-row softmax stats into slots [0, C)
__global__ void k_warmup(const float* __restrict__ W, const float* __restrict__ bias,
                         float* ents, float* confs, int* labs, int* valid, int C) {
  extern __shared__ float wl[];  // C*C
  int tid = threadIdx.x;
  for (int p = tid; p < C * C; p += blockDim.x) {
    int i = p / C, j = p % C;
    const float* wi = W + (long)i * D_DIM;
    const float* wj = W + (long)j * D_DIM;
    float s = 0.f;
    for (int k = 0; k < D_DIM; ++k) s += wi[k] * wj[k];
    wl[p] = s + bias[j];
  }
  __syncthreads();
  if (tid < C) {
    const float* row = wl + tid * C;
    float mx = row[0]; int am = 0;
    for (int j = 1; j < C; ++j) if (row[j] > mx) { mx = row[j]; am = j; }
    float se = 0.f, spl = 0.f;
    for (int j = 0; j < C; ++j) { float e = expf(row[j] - mx); se += e; spl += e * row[j]; }
    float lse = mx + logf(se);
    ents[tid]  = lse - spl / se;   // entropy
    confs[tid] = 1.0f / se;        // pmax = exp(mx - lse)
    labs[tid]  = am;
    valid[tid] = 1;
  }
}

// ---------------------------------------------------------------- stage B:
// logits = feat@W^T + b via WMMA f32 16x16x4; per-row entropy/argmax/pmax.
// grid = B/16 blocks, 128 threads (4 waves; wave w handles columns [16w,16w+16))
__global__ void k_logits(const float* __restrict__ feat, const float* __restrict__ W,
                         const float* __restrict__ bias,
                         float* ents_b, float* confs_b, int* labs_b, int C) {
  __shared__ float tile[16][64];
  int wave = threadIdx.x >> 5;
  int lane = threadIdx.x & 31;
  int lo   = lane & 15;
  int hi   = lane >> 4;                       // 0: K 0..1, 1: K 2..3
  long row0 = (long)blockIdx.x * 16;
  int col0 = wave * 16;
  int col  = col0 + lo;
  int colc = (col < C) ? col : (C - 1);       // clamp: pad cols never read back
  const float* arow = feat + (row0 + lo) * D_DIM + 2 * hi;
  const float* brow = W + (long)colc * D_DIM + 2 * hi;

  v8f acc = tile_gemm(arow, brow);

#pragma unroll
  for (int r = 0; r < 8; ++r) tile[r + 8 * hi][col0 + lo] = acc[r];
  __syncthreads();
  if (threadIdx.x < 16) {
    int m = threadIdx.x;
    long row = row0 + m;
    float mx = tile[m][0] + bias[0]; int am = 0;
    for (int j = 1; j < C; ++j) { float l = tile[m][j] + bias[j]; if (l > mx) { mx = l; am = j; } }
    float se = 0.f, spl = 0.f;
    for (int j = 0; j < C; ++j) { float l = tile[m][j] + bias[j]; float e = expf(l - mx); se += e; spl += e * l; }
    float lse = mx + logf(se);
    ents_b[row]  = lse - spl / se;
    confs_b[row] = 1.0f / se;
    labs_b[row]  = am;
  }
}

// ---------------------------------------------------------------- feat row L2 norms
// 256 threads = 8 waves; one wave per row
__global__ void k_rownorm(const float* __restrict__ feat, float* rnorm, int B) {
  int wave = threadIdx.x >> 5, lane = threadIdx.x & 31;
  long row = (long)blockIdx.x * 8 + wave;
  if (row >= B) return;
  const float* f = feat + row * D_DIM;
  float ss = 0.f;
  for (int k = lane; k < D_DIM; k += 32) { float v = f[k]; ss += v * v; }
#pragma unroll
  for (int off = 16; off > 0; off >>= 1) ss += __shfl_xor(ss, off, 32);
  if (lane == 0) rnorm[row] = sqrtf(ss);
}

// ---------------------------------------------------------------- stage C:
// single block: mean of ent -> dyn_q; bitonic sort of B entropies in dynamic LDS
// (256KB fits the 320KB/WGP CDNA5 LDS); exact interpolated quantile -> scal[0..1]
__global__ void k_sort_thresh(const float* __restrict__ ent, int N, float* scal) {
  extern __shared__ float sdata[];       // N floats
  __shared__ float red[1024];
  int tid = threadIdx.x;
  float ls = 0.f;
  for (int i = tid; i < N; i += 1024) { float v = ent[i]; sdata[i] = v; ls += v; }
  red[tid] = ls; __syncthreads();
  for (int s = 512; s > 0; s >>= 1) { if (tid < s) red[tid] += red[tid + s]; __syncthreads(); }
  float m = red[0] / (float)N;
  __syncthreads();
  for (unsigned k = 2; k <= (unsigned)N; k <<= 1) {
    for (unsigned j = k >> 1; j > 0; j >>= 1) {
      for (unsigned i = tid; i < (unsigned)N; i += 1024) {
        unsigned ixj = i ^ j;
        if (ixj > i) {
          bool up = ((i & k) == 0);
          float x = sdata[i], y = sdata[ixj];
          if ((x > y) == up) { sdata[i] = y; sdata[ixj] = x; }
        }
      }
      __syncthreads();
    }
  }
  if (tid == 0) {
    float q = (m >= 0.45f) ? 0.25f : ((m >= 0.38f) ? 0.30f : 0.40f);
    float pos = q * (float)(N - 1);
    int lo = (int)floorf(pos);
    int hi = min(lo + 1, N - 1);
    float frac = pos - (float)lo;
    scal[0] = sdata[lo] * (1.f - frac) + sdata[hi] * frac;   // thr
    scal[1] = (m >= 0.45f) ? 0.72f : 0.62f;                  // conf_thr
  }
}

// ---------------------------------------------------------------- stage D: mask
__global__ void k_mask(const float* __restrict__ lraw, const float* __restrict__ laug,
                       const float* __restrict__ ents_b, const float* __restrict__ confs_b,
                       int* valid_b, const float* __restrict__ scal, int B, int C) {
  long i = (long)blockIdx.x * blockDim.x + threadIdx.x;
  if (i >= B) return;
  const float* r = lraw + i * C;
  const float* a = laug + i * C;
  float mr = r[0]; int ar = 0; float ma = a[0]; int aa = 0;
  for (int j = 1; j < C; ++j) {
    if (r[j] > mr) { mr = r[j]; ar = j; }
    if (a[j] > ma) { ma = a[j]; aa = j; }
  }
  valid_b[i] = (ents_b[i] <= scal[0]) && (ar == aa) && (confs_b[i] >= scal[1]);
}

// ---------------------------------------------------------------- stage E:
// per-class top-10 by (ent-bits || idx) 64-bit key (stable-sort tie-break)
__global__ void k_topk(const float* __restrict__ ents, const int* __restrict__ labs,
                       const int* __restrict__ valid, int* topk, int N) {
  const unsigned long long MAXK = ~0ull;
  int c = blockIdx.x;
  int tid = threadIdx.x;
  unsigned long long loc[KMAX];
#pragma unroll
  for (int r = 0; r < KMAX; ++r) loc[r] = MAXK;
  for (int i = tid; i < N; i += blockDim.x) {
    if (labs[i] == c && valid[i]) {
      unsigned long long key =
          ((unsigned long long)__float_as_uint(ents[i]) << 32) | (unsigned)i;
      if (key < loc[KMAX - 1]) {
        int r = KMAX - 1;
        while (r > 0 && loc[r - 1] > key) { loc[r] = loc[r - 1]; --r; }
        loc[r] = key;
      }
    }
  }
  __shared__ unsigned long long pool[256 * KMAX];
  __shared__ unsigned long long red[256];
  for (int r = 0; r < KMAX; ++r) pool[tid * KMAX + r] = loc[r];
  __syncthreads();
  for (int r = 0; r < KMAX; ++r) {
    unsigned long long mn = MAXK;
    for (int p = tid; p < 256 * KMAX; p += blockDim.x) {
      unsigned long long v = pool[p];
      mn = (v < mn) ? v : mn;
    }
    red[tid] = mn; __syncthreads();
    for (int s = 128; s > 0; s >>= 1) {
      if (tid < s) { unsigned long long v = red[tid + s]; if (v < red[tid]) red[tid] = v; }
      __syncthreads();
    }
    unsigned long long w = red[0];
    __syncthreads();
    if (w != MAXK)
      for (int p = tid; p < 256 * KMAX; p += blockDim.x)
        if (pool[p] == w) pool[p] = MAXK;
    if (tid == 0) topk[c * KMAX + r] = (w == MAXK) ? -1 : (int)(unsigned)(w & 0xffffffffull);
    __syncthreads();
  }
}

// ---------------------------------------------------------------- stage F:
// centroid_c = normalize( sum_{kept e} max(conf_e,1e-6) * normalize(support_e) )
__global__ void k_centroid(const float* __restrict__ W, const float* __restrict__ feat,
                           const float* __restrict__ confs, const int* __restrict__ topk,
                           float* cent, int C) {
  __shared__ float accum[D_DIM];
  __shared__ float red[256];
  int c = blockIdx.x;
  int tid = threadIdx.x;
  for (int d = tid; d < D_DIM; d += blockDim.x) accum[d] = 0.f;
  __syncthreads();
  for (int r = 0; r < KMAX; ++r) {
    int idx = topk[c * KMAX + r];
    if (idx >= 0) {
      const float* v = (idx < C) ? (W + (long)idx * D_DIM)
                                 : (feat + (long)(idx - C) * D_DIM);
      float ss = 0.f;
      for (int d = tid; d < D_DIM; d += blockDim.x) ss += v[d] * v[d];
      red[tid] = ss; __syncthreads();
      for (int s = 128; s > 0; s >>= 1) { if (tid < s) red[tid] += red[tid + s]; __syncthreads(); }
      float nrm = sqrtf(red[0]);
      __syncthreads();
      float wgt = fmaxf(confs[idx], 1e-6f) / fmaxf(nrm, EPSV);
      for (int d = tid; d < D_DIM; d += blockDim.x) accum[d] += wgt * v[d];
      __syncthreads();
    }
  }
  float ss = 0.f;
  for (int d = tid; d < D_DIM; d += blockDim.x) ss += accum[d] * accum[d];
  red[tid] = ss; __syncthreads();
  for (int s = 128; s > 0; s >>= 1) { if (tid < s) red[tid] += red[tid + s]; __syncthreads(); }
  float nrm = fmaxf(sqrtf(red[0]), EPSV);
  for (int d = tid; d < D_DIM; d += blockDim.x) cent[(long)c * D_DIM + d] = accum[d] / nrm;
}

// ---------------------------------------------------------------- stage G:
// out = 20 * (feat @ cent^T) / ||feat_row||   via WMMA f32 16x16x4
__global__ void k_sim(const float* __restrict__ feat, const float* __restrict__ cent,
                      const float* __restrict__ rnorm, float* __restrict__ out, int C) {
  int wave = threadIdx.x >> 5;
  int lane = threadIdx.x & 31;
  int lo   = lane & 15;
  int hi   = lane >> 4;
  long row0 = (long)blockIdx.x * 16;
  int col0 = wave * 16;
  int col  = col0 + lo;
  int colc = (col < C) ? col : (C - 1);
  const float* arow = feat + (row0 + lo) * D_DIM + 2 * hi;
  const float* brow = cent + (long)colc * D_DIM + 2 * hi;

  v8f acc = tile_gemm(arow, brow);

  if (col < C) {
#pragma unroll
    for (int r = 0; r < 8; ++r) {
      long row = row0 + r + 8 * hi;
      float rn = fmaxf(rnorm[row], EPSV);
      out[row * C + col] = 20.0f * acc[r] / rn;
    }
  }
}

extern "C" void kernel_launch(void* const* d_in, const int* in_sizes, int n_in,
                              void* d_out, int out_size, void* d_ws, size_t ws_size,
                              hipStream_t stream) {
  (void)n_in; (void)out_size; (void)ws_size;
  const float* feat = (const float*)d_in[0];
  const float* lraw = (const float*)d_in[1];
  const float* laug = (const float*)d_in[2];
  const float* W    = (const float*)d_in[3];
  const float* bias = (const float*)d_in[4];
  float* out = (float*)d_out;

  const int D = D_DIM;
  const int C = in_sizes[4];            // 50
  const int B = in_sizes[0] / D;        // 65536
  const size_t NE = (size_t)C + (size_t)B;

  // workspace carve
  float* ents  = (float*)d_ws;                    // NE
  float* confs = ents + NE;                       // NE
  int*   labs  = (int*)(confs + NE);              // NE
  int*   valid = labs + NE;                       // NE
  float* rnorm = (float*)(valid + NE);            // B
  float* cent  = rnorm + B;                       // C*D
  int*   topk  = (int*)(cent + (size_t)C * D);    // C*KMAX
  float* scal  = (float*)(topk + C * KMAX);       // thr, conf_thr

  // A: warmup memory stats (slots [0,C))
  k_warmup<<<1, 256, (size_t)C * C * sizeof(float), stream>>>(W, bias, ents, confs, labs, valid, C);
  // B: logits GEMM (WMMA) + per-row stats (slots [C, C+B))
  k_logits<<<B / 16, 128, 0, stream>>>(feat, W, bias, ents + C, confs + C, labs + C, C);
  // feat row norms
  k_rownorm<<<(B + 7) / 8, 256, 0, stream>>>(feat, rnorm, B);
  // C: mean + bitonic sort in 256KB LDS -> thr, conf_thr
  k_sort_thresh<<<1, 1024, (size_t)B * sizeof(float), stream>>>(ents + C, B, scal);
  // D: acceptance mask for batch entries
  k_mask<<<(B + 255) / 256, 256, 0, stream>>>(lraw, laug, ents + C, confs + C, valid + C, scal, B, C);
  // E: per-class top-10 lowest-entropy (stable tie-break via 64-bit key)
  k_topk<<<C, 256, 0, stream>>>(ents, labs, valid, topk, (int)NE);
  // F: confidence-weighted normalized centroids
  k_centroid<<<C, 256, 0, stream>>>(W, feat, confs, topk, cent, C);
  // G: sim GEMM (WMMA), fold row-normalization + scale
  k_sim<<<B / 16, 128, 0, stream>>>(feat, cent, rnorm, out, C);
}